// MultiStepUnitaryGCN_53807350284449
// MI455X (gfx1250) — compile-verified
//
#include <hip/hip_runtime.h>
#include <hip/hip_bf16.h>
#include <math.h>

typedef __attribute__((ext_vector_type(2))) float v2f;
typedef __attribute__((ext_vector_type(8))) float v8f;

namespace {
constexpr int kN    = 50000;  // divisible by 16 -> no tile bounds checks
constexpr int kK    = 16;     // max degree
constexpr int kCIN  = 128;
constexpr int kCHID = 128;
constexpr int kCOUT = 64;
constexpr int kPAD  = 132;    // padded LDS row stride (floats): 132 % 64 = 4 -> conflict-free
}

// ---------------------------------------------------------------------------
// Kernel 0: per-node evolution scales.  Star adjacency A = e0 u^T + u e0^T,
// Re(expm(-itA)[0,:]) = [cos(t*sqrt(deg)), 0, ..., 0], norm == 1 (unitary).
// ---------------------------------------------------------------------------
__global__ __launch_bounds__(256) void scale_kernel(
    const unsigned char* __restrict__ mask, float* __restrict__ s1,
    float* __restrict__ s2, int n) {
  int i = blockIdx.x * blockDim.x + threadIdx.x;
  if (i >= n) return;
  int d = 0;
#pragma unroll
  for (int j = 0; j < kK; ++j) d += mask[i * kK + j] ? 1 : 0;
  float sq = sqrtf((float)d);
  s1[i] = cosf(1.0f * sq);   // T1
  s2[i] = cosf(0.5f * sq);   // T2
}

// ---------------------------------------------------------------------------
// Kernel 1: h = relu( s1 * (x @ W1^T) + b1 )
// Block: 256 threads = 8 waves; one 16-row node tile; wave w -> output cols
// [16w, 16w+16).  A (16x4 f32) per ISA layout: lanes 0-15 carry K={0,1},
// lanes 16-31 carry K={2,3}.  B (4x16) mirrored: lane's W-row fixed, K split.
// ---------------------------------------------------------------------------
__global__ __launch_bounds__(256) void l1_kernel(
    const float* __restrict__ x, const float* __restrict__ W1,
    const float* __restrict__ b1, const float* __restrict__ s1,
    float* __restrict__ h) {
  __shared__ float xs[16 * kPAD];
  __shared__ float ss[16];
  const int tid = threadIdx.x;
  const int rowBase = blockIdx.x * 16;

  // stage 16x128 x tile into LDS (padded rows), b128 per thread
  {
    int r = tid >> 4;                  // 0..15
    int c = (tid & 15) << 3;           // 0,8,...,120
    const float* src = x + (size_t)(rowBase + r) * kCIN + c;
    float* dst = xs + r * kPAD + c;
#pragma unroll
    for (int j = 0; j < 8; ++j) dst[j] = src[j];
  }
  if (tid < 16) ss[tid] = s1[rowBase + tid];
  __syncthreads();

  const int wave  = tid >> 5;
  const int lane  = tid & 31;
  const int nbase = wave * 16;            // output-channel tile base
  const int l16   = lane & 15;
  const int khalf = (lane >> 4) << 1;     // 0 for lanes 0-15, 2 for 16-31
  const float* xrow = xs + l16 * kPAD + khalf;                  // A row (M=l16)
  const float* wrow = W1 + (size_t)(nbase + l16) * kCIN + khalf; // B col (N=l16)

  v8f c = {0.f, 0.f, 0.f, 0.f, 0.f, 0.f, 0.f, 0.f};
#pragma unroll 8
  for (int k0 = 0; k0 < kCIN; k0 += 4) {
    v2f a = *(const v2f*)(xrow + k0);
    v2f b = *(const v2f*)(wrow + k0);
    c = __builtin_amdgcn_wmma_f32_16x16x4_f32(
        /*neg_a=*/false, a, /*neg_b=*/false, b,
        /*c_mod=*/(short)0, c, /*reuse_a=*/false, /*reuse_b=*/false);
  }

  // epilogue: D layout -> VGPR r, lanes 0-15: M=r; lanes 16-31: M=r+8; N=l16
  const int mlo = (lane < 16) ? 0 : 8;
  const int col = nbase + l16;
  const float bb = b1[col];
  float* hbase = h + (size_t)(rowBase + mlo) * kCHID + col;
#pragma unroll
  for (int r = 0; r < 8; ++r) {
    float v = c[r] * ss[r + mlo] + bb;
    hbase[(size_t)r * kCHID] = v > 0.f ? v : 0.f;
  }
}

// ---------------------------------------------------------------------------
// Kernel 2: o = s2 * (h @ W2^T) + b2 ; out = log_softmax(o, axis=1)
// Block: 128 threads = 4 waves (4 column tiles of 16 = 64 channels).
// ---------------------------------------------------------------------------
__global__ __launch_bounds__(128) void l2_kernel(
    const float* __restrict__ h, const float* __restrict__ W2,
    const float* __restrict__ b2, const float* __restrict__ s2,
    float* __restrict__ out) {
  __shared__ float hs[16 * kPAD];
  __shared__ float os[16 * kCOUT];
  __shared__ float ss[16];
  const int tid = threadIdx.x;
  const int rowBase = blockIdx.x * 16;

  // stage 16x128 h tile into LDS: 128 threads, 16 floats each
  {
    int r = tid >> 3;                  // 0..15
    int c = (tid & 7) << 4;            // 0,16,...,112
    const float* src = h + (size_t)(rowBase + r) * kCHID + c;
    float* dst = hs + r * kPAD + c;
#pragma unroll
    for (int j = 0; j < 16; ++j) dst[j] = src[j];
  }
  if (tid < 16) ss[tid] = s2[rowBase + tid];
  __syncthreads();

  const int wave  = tid >> 5;
  const int lane  = tid & 31;
  const int nbase = wave * 16;
  const int l16   = lane & 15;
  const int khalf = (lane >> 4) << 1;
  const float* hrow = hs + l16 * kPAD + khalf;
  const float* wrow = W2 + (size_t)(nbase + l16) * kCHID + khalf;

  v8f c = {0.f, 0.f, 0.f, 0.f, 0.f, 0.f, 0.f, 0.f};
#pragma unroll 8
  for (int k0 = 0; k0 < kCHID; k0 += 4) {
    v2f a = *(const v2f*)(hrow + k0);
    v2f b = *(const v2f*)(wrow + k0);
    c = __builtin_amdgcn_wmma_f32_16x16x4_f32(
        false, a, false, b, (short)0, c, false, false);
  }

  const int mlo = (lane < 16) ? 0 : 8;
  const int col = nbase + l16;
  const float bb = b2[col];
#pragma unroll
  for (int r = 0; r < 8; ++r) {
    os[(r + mlo) * kCOUT + col] = c[r] * ss[r + mlo] + bb;
  }
  __syncthreads();

  // per-row log_softmax over 64 channels (threads 0-15, one row each)
  if (tid < 16) {
    const float* row = os + tid * kCOUT;
    float mx = row[0];
#pragma unroll
    for (int j = 1; j < kCOUT; ++j) mx = fmaxf(mx, row[j]);
    float sum = 0.f;
#pragma unroll
    for (int j = 0; j < kCOUT; ++j) sum += expf(row[j] - mx);
    float lse = mx + logf(sum);
    float* orow = out + (size_t)(rowBase + tid) * kCOUT;
#pragma unroll
    for (int j = 0; j < kCOUT; ++j) orow[j] = row[j] - lse;
  }
}

// ---------------------------------------------------------------------------
// inputs: x[N*128] f32, neighbors[N*16] i32 (analytically unused: star-expm
// row-0 weights are exactly [cos(t*sqrt(deg)),0,...,0]), nbr_mask[N*16] bool,
// W1[128*128], b1[128], W2[64*128], b2[64]. out: [N*64] f32 log-probs.
// ---------------------------------------------------------------------------
extern "C" void kernel_launch(void* const* d_in, const int* in_sizes, int n_in,
                              void* d_out, int out_size, void* d_ws, size_t ws_size,
                              hipStream_t stream) {
  const float* x          = (const float*)d_in[0];
  const unsigned char* nm = (const unsigned char*)d_in[2];  // bool mask, 1B/elem
  const float* W1         = (const float*)d_in[3];
  const float* b1         = (const float*)d_in[4];
  const float* W2         = (const float*)d_in[5];
  const float* b2         = (const float*)d_in[6];
  float* out              = (float*)d_out;

  float* s1 = (float*)d_ws;            // N floats
  float* s2 = s1 + kN;                 // N floats
  float* h  = s2 + kN;                 // N*128 floats (~25.6 MB)

  scale_kernel<<<(kN + 255) / 256, 256, 0, stream>>>(nm, s1, s2, kN);
  l1_kernel<<<kN / 16, 256, 0, stream>>>(x, W1, b1, s1, h);
  l2_kernel<<<kN / 16, 128, 0, stream>>>(h, W2, b2, s2, out);
}